// LinearMaskInferenceOr_35424890257450
// MI455X (gfx1250) — compile-verified
//
#include <hip/hip_runtime.h>
#include <hip/hip_bf16.h>
#include <math.h>

// Problem constants from the reference: B=8, N=512, M=512, C=64, TAU=1, DROP=0.5
#define BB      8
#define NN      512
#define MM      512
#define CC      64
#define RTOT    (BB * NN * MM)        // 2,097,152 rows of 64 channels
#define KSEL    255                   // K-1, K = 512*0.5 = 256 (kth smallest, 0-indexed)
#define ROWSTR  68                    // LDS row stride in floats (pad: conflict-free, 16B aligned)

typedef float v2f __attribute__((ext_vector_type(2)));
typedef float v8f __attribute__((ext_vector_type(8)));

// ---------------------------------------------------------------------------
// Pass 1: s = sigmoid(dot(x_row, W) + bias + noise)  via V_WMMA_F32_16X16X4_F32
// A-matrix = W broadcast to all 16 rows (16x4 f32), B-matrix = 16 activation
// rows (4x16 f32).  D[m][n] = dot(X[n], W) for every m, so D VGPR0 carries all
// 16 results striped over lanes 0..15.
// Block = 256 threads = 8 waves; each wave owns a 16-row tile (4 KB), staged
// through LDS with fully coalesced float4 global loads.
// ---------------------------------------------------------------------------
__global__ void __launch_bounds__(256)
logits_sigmoid_wmma(const float* __restrict__ X,
                    const float* __restrict__ noise,
                    const float* __restrict__ W,
                    const float* __restrict__ bias,
                    float* __restrict__ S) {
  __shared__ __align__(16) float ldsX[8 * 16 * ROWSTR];  // 8 waves * 16 rows * 68
  __shared__ float ldsW[CC];

  const int tid  = threadIdx.x;
  const int lane = tid & 31;
  const int wave = tid >> 5;

  if (tid < CC) ldsW[tid] = W[tid];

  const long long blockRow0 = (long long)blockIdx.x * 128;     // 128 rows / block
  // Stage this wave's 16 rows (1024 contiguous floats) -> padded LDS region.
  float* wlds = ldsX + wave * (16 * ROWSTR);
  const float4* wsrc = (const float4*)(X + (blockRow0 + (long long)wave * 16) * CC);
#pragma unroll
  for (int it = 0; it < 8; ++it) {
    const int f   = (it * 32 + lane) * 4;   // flat float index 0..1020
    const float4 v = wsrc[it * 32 + lane];
    const int row = f >> 6;                 // / 64
    const int col = f & 63;
    *(float4*)(wlds + row * ROWSTR + col) = v;
  }
  __syncthreads();

  // WMMA accumulate over C=64 in 16 chunks of K=4.
  const int hi = lane >> 4;                 // lanes 16..31 carry K+2 / K+3
  const int n  = lane & 15;                 // B-matrix column = activation row
  const float* brow = wlds + n * ROWSTR;
  v8f acc = {0.f, 0.f, 0.f, 0.f, 0.f, 0.f, 0.f, 0.f};
#pragma unroll
  for (int kk = 0; kk < 16; ++kk) {
    const int kb = kk * 4 + (hi ? 2 : 0);
    v2f a, bm;
    a.x  = ldsW[kb];
    a.y  = ldsW[kb + 1];
    bm.x = brow[kb];
    bm.y = brow[kb + 1];
    acc = __builtin_amdgcn_wmma_f32_16x16x4_f32(
        /*neg_a=*/false, a, /*neg_b=*/false, bm,
        /*c_mod=*/(short)0, acc, /*reuse_a=*/false, /*reuse_b=*/false);
  }

  // D VGPR0: lanes 0..15 hold dot(X[n], W); lanes 16..31 duplicate them.
  const float dotv = acc[0];
  if (lane < 16) {
    const long long r = blockRow0 + (long long)wave * 16 + n;
    const float z = dotv + bias[0] + noise[r];        // TAU == 1
    S[r] = 1.0f / (1.0f + expf(-z));
  }
}

// ---------------------------------------------------------------------------
// Pass 2: k-th smallest (index 255 of 512) per column, LDS bitonic sort.
// strideIsOne==1 : columns are contiguous 512-float runs (threshold over M)
// strideIsOne==0 : columns strided by 512 floats    (threshold over N)
// ---------------------------------------------------------------------------
__global__ void __launch_bounds__(256)
kth_select(const float* __restrict__ S, float* __restrict__ kth, int strideIsOne) {
  __shared__ float sd[NN];
  const int g   = blockIdx.x;               // 4096 columns
  const int tid = threadIdx.x;

  long long base;
  int stride;
  if (strideIsOne) { base = (long long)g * MM;                        stride = 1;  }
  else             { base = (long long)(g >> 9) * (NN * MM) + (g & 511); stride = MM; }

  for (int i = tid; i < NN; i += 256) sd[i] = S[base + (long long)i * stride];
  __syncthreads();

  for (int k = 2; k <= NN; k <<= 1) {
    for (int j = k >> 1; j > 0; j >>= 1) {
      for (int idx = tid; idx < NN; idx += 256) {
        const int l = idx ^ j;
        if (l > idx) {
          const float a = sd[idx];
          const float b = sd[l];
          const bool asc = ((idx & k) == 0);
          if ((a > b) == asc) { sd[idx] = b; sd[l] = a; }
        }
      }
      __syncthreads();
    }
  }
  if (tid == 0) kth[g] = sd[KSEL];
}

// ---------------------------------------------------------------------------
// Pass 3: straight-through masks, OR-combine with ==2.0 rescale (in place).
// ---------------------------------------------------------------------------
__global__ void __launch_bounds__(256)
finalize_or(float* __restrict__ out, const float* __restrict__ Sb,
            const float* __restrict__ kthA, const float* __restrict__ kthB) {
  const long long t = (long long)blockIdx.x * blockDim.x + threadIdx.x;
  const int m = (int)(t & 511);
  const int n = (int)((t >> 9) & 511);
  const int b = (int)(t >> 18);

  const float sa = out[t];
  const float sb = Sb[t];
  const float ka = kthA[b * MM + m];   // threshold over N, per (b, m)
  const float kb = kthB[b * NN + n];   // threshold over M, per (b, n)

  const float ha = (sa >= ka) ? 1.0f : 0.0f;
  const float hb = (sb >= kb) ? 1.0f : 0.0f;
  // straight-through: y = y_hard - stop_grad(y_soft) + y_soft (match fp order)
  const float va = (ha - sa) + sa;
  const float vb = (hb - sb) + sb;
  const float y  = va + vb;
  out[t] = (y == 2.0f) ? (y * 0.5f) : y;
}

extern "C" void kernel_launch(void* const* d_in, const int* in_sizes, int n_in,
                              void* d_out, int out_size, void* d_ws, size_t ws_size,
                              hipStream_t stream) {
  (void)in_sizes; (void)n_in; (void)out_size; (void)ws_size;
  const float* xab  = (const float*)d_in[0];
  const float* xba  = (const float*)d_in[1];
  const float* nab  = (const float*)d_in[2];
  const float* nba  = (const float*)d_in[3];
  const float* W    = (const float*)d_in[4];
  const float* bias = (const float*)d_in[5];

  float* out  = (float*)d_out;                 // s_a lives here, finalized in place
  float* sB   = (float*)d_ws;                  // RTOT floats (8 MB)
  float* kthA = sB + RTOT;                     // 4096 floats
  float* kthB = kthA + BB * MM;                // 4096 floats

  // Pass 1: 16384 blocks x 256 thr, 128 rows/block -> exactly RTOT rows.
  logits_sigmoid_wmma<<<dim3(RTOT / 128), dim3(256), 0, stream>>>(xab, nab, W, bias, out);
  logits_sigmoid_wmma<<<dim3(RTOT / 128), dim3(256), 0, stream>>>(xba, nba, W, bias, sB);

  // Pass 2: 4096 columns each.
  kth_select<<<dim3(BB * MM), dim3(256), 0, stream>>>(out, kthA, /*strideIsOne=*/0);
  kth_select<<<dim3(BB * NN), dim3(256), 0, stream>>>(sB,  kthB, /*strideIsOne=*/1);

  // Pass 3: elementwise combine.
  finalize_or<<<dim3(RTOT / 256), dim3(256), 0, stream>>>(out, sB, kthA, kthB);
}